// Attention_48945447305302
// MI455X (gfx1250) — compile-verified
//
#include <hip/hip_runtime.h>

typedef __attribute__((ext_vector_type(2))) float v2f;
typedef __attribute__((ext_vector_type(8))) float v8f;
typedef int i4v __attribute__((vector_size(16)));   // matches builtin's int4 pointee

#define B_ 64
#define T_ 1024
#define H_ 1024

#define BM 64
#define BN 128
#define BK 32
#define KP (BK + 4)   // 36-float padded LDS stride: banks (k + 36*m)%64 conflict-free

#define AS1 __attribute__((address_space(1)))
#define AS3 __attribute__((address_space(3)))

#if defined(__has_builtin)
#if __has_builtin(__builtin_amdgcn_global_load_async_to_lds_b128)
#define HAVE_ASYNC_LDS 1
#endif
#endif
#ifndef HAVE_ASYNC_LDS
#define HAVE_ASYNC_LDS 0
#endif

__device__ __forceinline__ void async_b128(const float* g, float* l) {
#if HAVE_ASYNC_LDS
  __builtin_amdgcn_global_load_async_to_lds_b128((AS1 i4v*)g, (AS3 i4v*)l, 0, 0);
#else
  *(float4*)l = *(const float4*)g;   // sync fallback
#endif
}

__device__ __forceinline__ void wait_async_lds() {
#if HAVE_ASYNC_LDS
#if __has_builtin(__builtin_amdgcn_s_wait_asynccnt)
  __builtin_amdgcn_s_wait_asynccnt(0);
#else
  asm volatile("s_wait_asynccnt 0x0" ::: "memory");
#endif
#endif
}

// ---------------- Kernel 1: dec[b,o] = hidden[b,:] . W_dec[o,:] + b_dec[o] ----
__global__ __launch_bounds__(256)
void dec_proj_kernel(const float* __restrict__ hidden,
                     const float* __restrict__ W_dec,
                     const float* __restrict__ b_dec,
                     float* __restrict__ dec) {
  __shared__ float hid[H_];
  const int b = blockIdx.x;
  const int tid = threadIdx.x;
  for (int i = tid; i < H_; i += 256) hid[i] = hidden[b * H_ + i];
  __syncthreads();
  #pragma unroll
  for (int j = 0; j < 4; ++j) {
    const int o = tid + j * 256;
    const float4* wr = (const float4*)(W_dec + (size_t)o * H_);
    const float4* hr = (const float4*)hid;
    float acc = 0.f;
    #pragma unroll 4
    for (int h4 = 0; h4 < H_ / 4; ++h4) {
      float4 wv = wr[h4];
      float4 hv = hr[h4];
      acc += wv.x * hv.x + wv.y * hv.y + wv.z * hv.z + wv.w * hv.w;
    }
    dec[b * H_ + o] = acc + b_dec[o];
  }
}

// ---------------- Kernel 2: fused enc-GEMM + tanh + score dot ----------------
// scores[b,t] = sum_o tanh( X[b,t,:].W_enc[o,:] + b_enc[o] + dec[b,o] ) * w_score[o] + b_score
__global__ __launch_bounds__(256)
void enc_score_kernel(const float* __restrict__ X,       // [B*T, H]
                      const float* __restrict__ W_enc,   // [H, H]
                      const float* __restrict__ b_enc,   // [H]
                      const float* __restrict__ dec,     // [B, H]
                      const float* __restrict__ w_score, // [H]
                      const float* __restrict__ b_score, // [1]
                      float* __restrict__ scores)        // [B, T]
{
  __shared__ float Xs[2][BM * KP];
  __shared__ float Ws[2][BN * KP];
  __shared__ float scorePart[8][16];

  const int tid  = threadIdx.x;
  const int lane = tid & 31;
  const int w    = tid >> 5;   // wave id 0..7
  const int mi   = w & 3;      // 16-row subtile within BM
  const int ni   = w >> 2;     // 64-col half within BN
  const int lm   = lane & 15;  // lane within 16-group
  const int hi   = lane >> 4;  // half-wave id (K split for f32 WMMA frags)

  const int r0 = blockIdx.x * BM;   // flat row = b*T + t  (BM divides T)
  const int b  = r0 / T_;
  const int t0 = r0 % T_;

  const float* Xrow = X + (size_t)r0 * H_;

  // stage one BMxBK (X) + BNxBK (W) chunk into LDS buffer `buf` (async if available)
  auto stage = [&](int n0, int k0, int buf) {
    float* xs = &Xs[buf][0];
    float* ws = &Ws[buf][0];
    // X: 64 rows x 8 float4 = 512 -> 2 per thread
    #pragma unroll
    for (int i = 0; i < 2; ++i) {
      int idx = tid + i * 256;
      int row = idx >> 3, c4 = idx & 7;
      async_b128(Xrow + (size_t)row * H_ + k0 + c4 * 4, &xs[row * KP + c4 * 4]);
    }
    // W: 128 rows x 8 float4 = 1024 -> 4 per thread
    #pragma unroll
    for (int i = 0; i < 4; ++i) {
      int idx = tid + i * 256;
      int row = idx >> 3, c4 = idx & 7;
      async_b128(W_enc + (size_t)(n0 + row) * H_ + k0 + c4 * 4, &ws[row * KP + c4 * 4]);
    }
  };

  float sp[8];
  #pragma unroll
  for (int r = 0; r < 8; ++r) sp[r] = 0.f;

  int cur = 0;
  stage(0, 0, 0);   // prologue

  for (int n0 = 0; n0 < H_; n0 += BN) {
    v8f acc[4];
    #pragma unroll
    for (int j = 0; j < 4; ++j) { v8f z = {}; acc[j] = z; }

    for (int k0 = 0; k0 < H_; k0 += BK) {
      wait_async_lds();    // own async stores to LDS complete
      __syncthreads();     // everyone's complete; buf[cur^1] readers done too

      // kick off the next chunk into the other buffer
      int nk = k0 + BK, nn = n0;
      if (nk == H_) { nk = 0; nn = n0 + BN; }
      if (nn < H_) stage(nn, nk, cur ^ 1);

      // A frag: lane holds X[mi*16+lm][k + hi*2 + {0,1}]  (ISA 16x4 f32 layout)
      const float* ap = &Xs[cur][(mi * 16 + lm) * KP + hi * 2];
      #pragma unroll
      for (int kk = 0; kk < BK / 4; ++kk) {
        v2f a = *(const v2f*)(ap + kk * 4);
        #pragma unroll
        for (int j = 0; j < 4; ++j) {
          // B frag: lane holds W_enc[n0+ni*64+j*16+lm][k + hi*2 + {0,1}]
          v2f bv = *(const v2f*)(&Ws[cur][(ni * 64 + j * 16 + lm) * KP + hi * 2] + kk * 4);
          acc[j] = __builtin_amdgcn_wmma_f32_16x16x4_f32(
              false, a, false, bv, (short)0, acc[j], false, false);
        }
      }
      cur ^= 1;
    }

    // fused epilogue: bias + tanh + score-dot partials (row = mi*16 + r + hi*8)
    #pragma unroll
    for (int j = 0; j < 4; ++j) {
      const int gn = n0 + ni * 64 + j * 16 + lm;
      const float bias = b_enc[gn] + dec[b * H_ + gn];
      const float wsc  = w_score[gn];
      #pragma unroll
      for (int r = 0; r < 8; ++r) {
        float e = tanhf(acc[j][r] + bias);
        sp[r] += e * wsc;
      }
    }
  }

  // deterministic reduction: within 16-lane groups, then across the two n-halves
  #pragma unroll
  for (int r = 0; r < 8; ++r) {
    float v = sp[r];
    v += __shfl_xor(v, 8, 16);
    v += __shfl_xor(v, 4, 16);
    v += __shfl_xor(v, 2, 16);
    v += __shfl_xor(v, 1, 16);
    if (lm == 0) scorePart[w][hi * 8 + r] = v;  // lanes 0 and 16 write
  }
  __syncthreads();
  if (tid < BM) {
    const int mt = tid >> 4, rr = tid & 15;
    float s = scorePart[mt][rr] + scorePart[mt + 4][rr] + b_score[0];
    scores[(size_t)b * T_ + t0 + tid] = s;
  }
}

// ---------------- Kernel 3: in-place softmax over T per batch row ------------
__global__ __launch_bounds__(256)
void softmax_kernel(float* __restrict__ scores) {
  __shared__ float red[256];
  const int b = blockIdx.x, tid = threadIdx.x;
  float* row = scores + (size_t)b * T_;
  float s0 = row[tid], s1 = row[tid + 256], s2 = row[tid + 512], s3 = row[tid + 768];
  float m = fmaxf(fmaxf(s0, s1), fmaxf(s2, s3));
  red[tid] = m;
  __syncthreads();
  for (int off = 128; off > 0; off >>= 1) {
    if (tid < off) red[tid] = fmaxf(red[tid], red[tid + off]);
    __syncthreads();
  }
  m = red[0];
  __syncthreads();
  float e0 = expf(s0 - m), e1 = expf(s1 - m), e2 = expf(s2 - m), e3 = expf(s3 - m);
  red[tid] = e0 + e1 + e2 + e3;
  __syncthreads();
  for (int off = 128; off > 0; off >>= 1) {
    if (tid < off) red[tid] += red[tid + off];
    __syncthreads();
  }
  const float inv = 1.f / red[0];
  row[tid] = e0 * inv;
  row[tid + 256] = e1 * inv;
  row[tid + 512] = e2 * inv;
  row[tid + 768] = e3 * inv;
}

// ---------------- Kernel 4: context[b,h] = sum_t X[b,t,h] * w[b,t] -----------
// grid (H/256, B); 4-way t-split per block, float4 loads, deterministic combine.
__global__ __launch_bounds__(256)
void context_kernel(const float* __restrict__ X,
                    const float* __restrict__ weights,
                    float* __restrict__ context) {
  __shared__ float wl[T_];
  __shared__ float part[4][256];
  const int b   = blockIdx.y;
  const int tid = threadIdx.x;
  const int tg  = tid >> 6;          // t-group 0..3 (256 t each)
  const int hl  = (tid & 63) * 4;    // local h (float4 aligned)
  const int h0  = blockIdx.x * 256;
  for (int i = tid; i < T_; i += 256) wl[i] = weights[(size_t)b * T_ + i];
  __syncthreads();
  const float* xb = X + (size_t)b * T_ * H_ + h0 + hl;
  float4 acc = make_float4(0.f, 0.f, 0.f, 0.f);
  #pragma unroll 4
  for (int t = tg * 256; t < tg * 256 + 256; ++t) {
    float4 xv = *(const float4*)(xb + (size_t)t * H_);
    float wv = wl[t];
    acc.x += xv.x * wv; acc.y += xv.y * wv;
    acc.z += xv.z * wv; acc.w += xv.w * wv;
  }
  *(float4*)&part[tg][hl] = acc;
  __syncthreads();
  float s = part[0][tid] + part[1][tid] + part[2][tid] + part[3][tid];
  context[(size_t)b * H_ + h0 + tid] = s;
}

// ---------------- launch -----------------------------------------------------
extern "C" void kernel_launch(void* const* d_in, const int* in_sizes, int n_in,
                              void* d_out, int out_size, void* d_ws, size_t ws_size,
                              hipStream_t stream) {
  const float* X       = (const float*)d_in[0];  // spatial_feats [B,T,H]
  const float* hidden  = (const float*)d_in[1];  // hidden_state  [B,H]
  const float* W_enc   = (const float*)d_in[2];  // [H,H]
  const float* b_enc   = (const float*)d_in[3];  // [H]
  const float* W_dec   = (const float*)d_in[4];  // [H,H]
  const float* b_dec   = (const float*)d_in[5];  // [H]
  const float* w_score = (const float*)d_in[6];  // [H]
  const float* b_score = (const float*)d_in[7];  // scalar

  float* context = (float*)d_out;                   // output 0: [B,H]
  float* weights = (float*)d_out + (size_t)B_ * H_; // output 1: [B,T]
  float* dec     = (float*)d_ws;                    // scratch: [B,H]

  dec_proj_kernel<<<B_, 256, 0, stream>>>(hidden, W_dec, b_dec, dec);
  enc_score_kernel<<<(B_ * T_) / BM, 256, 0, stream>>>(X, W_enc, b_enc, dec,
                                                       w_score, b_score, weights);
  softmax_kernel<<<B_, 256, 0, stream>>>(weights);
  context_kernel<<<dim3(H_ / 256, B_), 256, 0, stream>>>(X, weights, context);
}